// FullGroupConv2D_76373108457475
// MI455X (gfx1250) — compile-verified
//
#include <hip/hip_runtime.h>
#include <hip/hip_bf16.h>

// Depthwise 3x3 conv, pad=1, stride=1, NCHW, fp32.
// x: (16, 384, 128, 128), k: (384, 1, 3, 3) -> out: (16, 384, 128, 128)
// Memory-bound (2.25 flop/byte). Strategy: async global->LDS staging
// (CDNA5 GLOBAL_LOAD_ASYNC_TO_LDS_B128), compute from LDS, coalesced B128 stores.

#define N_BATCH 16
#define C_CH    384
#define H_IMG   128
#define W_IMG   128

#define TILE_ROWS 32
#define LDS_ROWS  (TILE_ROWS + 2)   // 34: tile + top/bottom halo rows
#define LDS_COLS  132               // cols 0..127 = x data, 128..131 = zero pad (right halo + left-halo alias)

__device__ __forceinline__ unsigned lds_off(const void* p) {
    // Generic pointers into __shared__ carry the DS byte offset in their low 32 bits
    // (flat LDS aperture = {aperture_hi32, ds_addr_lo32}).
    return (unsigned)(unsigned long long)p;
}

__global__ __launch_bounds__(256) void dwconv3x3_kernel(
    const float* __restrict__ x, const float* __restrict__ k, float* __restrict__ out)
{
    __shared__ __align__(16) float smem[LDS_ROWS * LDS_COLS];

    const int tile  = blockIdx.x & 3;       // 4 row-tiles of 32 rows
    const int plane = blockIdx.x >> 2;      // n*C + c
    const int c     = plane % C_CH;
    const int y0    = tile * TILE_ROWS;

    const int tid  = threadIdx.x;
    const int lane = tid & 31;
    const int wave = tid >> 5;

    const float* xplane = x   + (size_t)plane * (H_IMG * W_IMG);
    float*       oplane = out + (size_t)plane * (H_IMG * W_IMG);

    // ---- zero-fill LDS halo regions (disjoint from async-load targets) ----
    // right-side pad cols 128..131 of every row (also serves lane-0 left-halo read at col 131)
    if (tid < LDS_ROWS * 4) {
        int r  = tid >> 2;
        int cc = 128 + (tid & 3);
        smem[r * LDS_COLS + cc] = 0.0f;
    }
    // top / bottom halo rows at image boundary (these rows are skipped by the async loads)
    if (y0 == 0 && tid < LDS_COLS) smem[tid] = 0.0f;
    if (y0 + TILE_ROWS == H_IMG && tid < LDS_COLS)
        smem[(LDS_ROWS - 1) * LDS_COLS + tid] = 0.0f;

    // ---- async stage: 34 rows, one 512B row per wave per instruction ----
    // lane moves 16B: global (plane + yin*512 + lane*16) -> LDS (row*528 + lane*16), both 16B aligned.
    const unsigned lds_base = lds_off(smem);
    for (int r = wave; r < LDS_ROWS; r += 8) {
        const int yin = y0 - 1 + r;               // wave-uniform
        if (yin >= 0 && yin < H_IMG) {
            unsigned laddr = lds_base + (unsigned)((r * LDS_COLS + lane * 4) * 4);
            unsigned voff  = (unsigned)(yin * (W_IMG * 4) + lane * 16);
            asm volatile("global_load_async_to_lds_b128 %0, %1, %2"
                         :: "v"(laddr), "v"(voff), "s"(xplane)
                         : "memory");
        }
    }
    asm volatile("s_wait_asynccnt 0" ::: "memory");
    __syncthreads();

    // ---- per-channel 3x3 weights (block-uniform -> scalar loads) ----
    const float* kc = k + c * 9;
    const float kw0 = kc[0], kw1 = kc[1], kw2 = kc[2];
    const float kw3 = kc[3], kw4 = kc[4], kw5 = kc[5];
    const float kw6 = kc[6], kw7 = kc[7], kw8 = kc[8];

    // ---- compute: wave w -> output rows 4w..4w+3; lane -> cols lane*4..lane*4+3 ----
    const int x0 = lane * 4;
    const int li = (lane == 0) ? 131 : (x0 - 1);   // left neighbor (col 131 is zero)

    #pragma unroll
    for (int jj = 0; jj < 4; ++jj) {
        const int j = wave * 4 + jj;               // output row within tile
        float a0 = 0.f, a1 = 0.f, a2 = 0.f, a3 = 0.f;

        #pragma unroll
        for (int rr = 0; rr < 3; ++rr) {
            const float* row = &smem[(j + rr) * LDS_COLS];
            const float4 A = *(const float4*)(row + x0);       // x0..x0+3   (ds_load_b128)
            const float4 B = *(const float4*)(row + x0 + 4);   // x0+4..x0+7 (ds_load_b128)
            const float  left = row[li];                       // x0-1       (ds_load_b32)

            const float ka = (rr == 0) ? kw0 : (rr == 1) ? kw3 : kw6;
            const float kb = (rr == 0) ? kw1 : (rr == 1) ? kw4 : kw7;
            const float kq = (rr == 0) ? kw2 : (rr == 1) ? kw5 : kw8;

            a0 = fmaf(ka, left, fmaf(kb, A.x, fmaf(kq, A.y, a0)));
            a1 = fmaf(ka, A.x,  fmaf(kb, A.y, fmaf(kq, A.z, a1)));
            a2 = fmaf(ka, A.y,  fmaf(kb, A.z, fmaf(kq, A.w, a2)));
            a3 = fmaf(ka, A.z,  fmaf(kb, A.w, fmaf(kq, B.x, a3)));
        }

        float4 o; o.x = a0; o.y = a1; o.z = a2; o.w = a3;
        *(float4*)(oplane + (size_t)(y0 + j) * W_IMG + x0) = o;   // global_store_b128
    }
}

extern "C" void kernel_launch(void* const* d_in, const int* in_sizes, int n_in,
                              void* d_out, int out_size, void* d_ws, size_t ws_size,
                              hipStream_t stream) {
    const float* x = (const float*)d_in[0];
    const float* k = (const float*)d_in[1];
    float* out = (float*)d_out;

    const int planes = N_BATCH * C_CH;                 // 6144
    const int tiles_per_plane = H_IMG / TILE_ROWS;     // 4
    dim3 grid(planes * tiles_per_plane);               // 24576 blocks
    dim3 block(256);                                   // 8 wave32

    dwconv3x3_kernel<<<grid, block, 0, stream>>>(x, k, out);
}